// FullPrecisionAttentionBlock_87290915324055
// MI455X (gfx1250) — compile-verified
//
#include <hip/hip_runtime.h>
#include <math.h>

typedef __attribute__((ext_vector_type(2))) float v2f;
typedef __attribute__((ext_vector_type(8))) float v8f;

#define EPS 1e-5f
#define B_  4
#define N_  256
#define DE_ 128
#define DV_ 64

// d_out offsets (in floats), return order: est_latent, out, Qij, Zhat, lambda_h
#define OFF_EST   0
#define OFF_OUT   131072
#define OFF_QIJ   393216
#define OFF_ZHAT  17170432
#define OFF_LAM   50724864

// ws layout (floats): Qr @0 (65536), Qi @65536, packed KV float4 @131072
// kv[(b*256+n)*64 + d] = {Kr, Ki, Vr, Vi}  -> 262144 floats. Total 1.5 MB.
#define WS_QI  65536
#define WS_KV  131072

// ---------------------------------------------------------------------------
// Kernel 1: complex projections Q,K,V via V_WMMA_F32_16X16X4_F32.
// One wave computes a 16x16 tile of one (yr,yi) pair; 4 f32 accumulators.
// grid = 96 blocks * 8 waves = 768 waves = 3 proj * 64 Mtiles * 4 Ntiles.
// K/V results are stored interleaved (float4 per (row,d)) so the attention
// kernel can fetch all four operands with a single global_load_b128.
// ---------------------------------------------------------------------------
__global__ __launch_bounds__(256) void proj_qkv_kernel(
    const float* __restrict__ Zq, const float* __restrict__ Zk, const float* __restrict__ Zv,
    const float* __restrict__ Wqr, const float* __restrict__ Wqi,
    const float* __restrict__ bqr, const float* __restrict__ bqi,
    const float* __restrict__ Wkr, const float* __restrict__ Wki,
    const float* __restrict__ bkr, const float* __restrict__ bki,
    const float* __restrict__ Wvr, const float* __restrict__ Wvi,
    const float* __restrict__ bvr, const float* __restrict__ bvi,
    const float* __restrict__ lam1,
    float* __restrict__ ws, float* __restrict__ out)
{
    // lambda_h output: re = -lam1[0]^2 (repeated), im = [lam1[1], -lam1[1]]
    if (blockIdx.x == 0 && threadIdx.x < 128) {
        int t = threadIdx.x;
        int d = t & 63;
        float l0 = lam1[d & 31];
        float re = -l0 * l0;
        float l1 = lam1[32 + (d & 31)];
        float im = (d < 32) ? l1 : -l1;
        out[OFF_LAM + t] = (t < 64) ? re : im;
    }

    int wave = threadIdx.x >> 5;
    int lane = threadIdx.x & 31;
    int gid  = blockIdx.x * 8 + wave;
    int p    = gid >> 8;          // projection 0=Q 1=K 2=V
    int rem  = gid & 255;
    int mt   = rem >> 2;          // 0..63  (rows of 1024)
    int nt   = rem & 3;           // 0..3   (cols of 64)

    const float *X, *Wr, *Wi, *br, *bi;
    if (p == 0)      { X=Zq; Wr=Wqr; Wi=Wqi; br=bqr; bi=bqi; }
    else if (p == 1) { X=Zk; Wr=Wkr; Wi=Wki; br=bkr; bi=bki; }
    else             { X=Zv; Wr=Wvr; Wi=Wvi; br=bvr; bi=bvi; }

    int mloc  = lane & 15;
    int khalf = lane >> 4;
    int row_m = mt * 16 + mloc;
    int rb    = row_m >> 8;              // batch
    int rn    = row_m & 255;             // seq pos
    int xr_base = (rb * 512 + rn) * DE_; // Z[b,0,n,:]
    int xi_base = xr_base + 256 * DE_;   // Z[b,1,n,:]
    int col = nt * 16 + mloc;

    v8f arr = {}; v8f aii = {}; v8f ari = {}; v8f air = {};
    for (int k = 0; k < DE_; k += 4) {
        int k0 = k + 2 * khalf;
        v2f ar, ai, bwr, bwi;
        ar.x  = X[xr_base + k0];      ar.y  = X[xr_base + k0 + 1];
        ai.x  = X[xi_base + k0];      ai.y  = X[xi_base + k0 + 1];
        bwr.x = Wr[k0 * DV_ + col];   bwr.y = Wr[(k0 + 1) * DV_ + col];
        bwi.x = Wi[k0 * DV_ + col];   bwi.y = Wi[(k0 + 1) * DV_ + col];
        arr = __builtin_amdgcn_wmma_f32_16x16x4_f32(false, ar, false, bwr, (short)0, arr, false, false);
        aii = __builtin_amdgcn_wmma_f32_16x16x4_f32(false, ai, false, bwi, (short)0, aii, false, false);
        ari = __builtin_amdgcn_wmma_f32_16x16x4_f32(false, ar, false, bwi, (short)0, ari, false, false);
        air = __builtin_amdgcn_wmma_f32_16x16x4_f32(false, ai, false, bwr, (short)0, air, false, false);
    }
    float brv = br[col], biv = bi[col];
#pragma unroll
    for (int v = 0; v < 8; ++v) {
        int row = mt * 16 + v + 8 * khalf;
        float yr = arr[v] - aii[v] + brv;   // yr = rr - ii + br
        float yi = ari[v] + air[v] + biv;   // yi = ri + ir + bi
        if (p == 0) {
            ws[row * DV_ + col]         = yr;
            ws[WS_QI + row * DV_ + col] = yi;
        } else {
            // packed {Kr,Ki,Vr,Vi}: K -> components 0,1 ; V -> components 2,3
            int base = WS_KV + (row * DV_ + col) * 4 + ((p == 1) ? 0 : 2);
            ws[base]     = yr;
            ws[base + 1] = yi;
        }
    }
}

// ---------------------------------------------------------------------------
// Kernel 2: fused attention body. One block per (b,i); scores tile in LDS
// (64 KB -- only viable with CDNA5's 320 KB WGP LDS).
// Writes Zhat, Qij, est_latent. 256 threads: d = t&63, j-group = t>>6.
// ---------------------------------------------------------------------------
__global__ __launch_bounds__(256) void attn_kernel(
    const float* __restrict__ T, const float* __restrict__ lam1,
    const float* __restrict__ lOm, const float* __restrict__ lGa,
    const float* __restrict__ nf_p, const float* __restrict__ tau_p,
    const float* __restrict__ nu_p, const float* __restrict__ eta_p,
    const float* __restrict__ ws, float* __restrict__ out)
{
    extern __shared__ float smem[];
    float* sc   = smem;              // N*DV  : Vij then scores (in place)
    float* dist = sc + N_ * DV_;     // N
    float* dmax = dist + N_;         // DV
    float* dsum = dmax + DV_;        // DV
    float* estv = dsum + DV_;        // 2*DV

    const float*  Qr  = ws;
    const float*  Qi  = ws + WS_QI;
    const float4* kv4 = (const float4*)(ws + WS_KV);

    int b = blockIdx.x >> 8;
    int i = blockIdx.x & 255;
    int t = threadIdx.x;
    int d = t & 63;
    int jg = t >> 6;

    float l0 = lam1[d & 31];
    float re = -l0 * l0;
    float l1 = lam1[32 + (d & 31)];
    float im = (d < 32) ? l1 : -l1;
    float lo = lOm[d]; lo = lo * lo;
    float lg = lGa[d]; lg = lg * lg + EPS;
    float ti = T[i];
    float nf      = nf_p[0];
    float tau_abs = fabsf(tau_p[0]);
    float nu_abs  = fabsf(nu_p[0]);
    int   rowQ = (b * 256 + i) * DV_ + d;
    float q_r = Qr[rowQ], q_i = Qi[rowQ];

    dist[t] = 0.0f;
    if (t < 2 * DV_) estv[t] = 0.0f;
    __syncthreads();

    size_t zbase_r = (size_t)OFF_ZHAT + ((size_t)(b * 2 + 0) * N_ + i) * N_ * DV_;
    size_t zbase_i = (size_t)OFF_ZHAT + ((size_t)(b * 2 + 1) * N_ + i) * N_ * DV_;
    size_t qbase   = (size_t)OFF_QIJ  + (size_t)(b * N_ + i) * N_ * DV_;

    // Stage A: Zhat, Vij stash, mahalanobis partial sums; zeros above diagonal
    for (int j = jg; j < N_; j += 4) {
        if (j <= i) {
            float dt = ti - T[j];
            float amp = __expf(re * dt);
            float s, c; __sincosf(im * dt, &s, &c);
            float Ker = amp * c, Kei = amp * s;
            float4 kv = kv4[(b * 256 + j) * DV_ + d];   // one global_load_b128
            float kr = kv.x, ki = kv.y, vr = kv.z, vi = kv.w;
            out[zbase_r + (size_t)j * DV_ + d] = Ker * vr - Kei * vi;  // re-read in stage D (L2)
            out[zbase_i + (size_t)j * DV_ + d] = Ker * vi + Kei * vr;
            float rr = q_r - (Ker * kr - Kei * ki);
            float ri = q_i - (Ker * ki + Kei * kr);
            float e2 = __expf(2.0f * re * fabsf(dt));
            float Vij = lg * e2 + lo * (e2 - 1.0f) / (2.0f * (re - EPS));
            sc[j * DV_ + d] = Vij;
            float mah = (rr * rr + ri * ri) / (Vij + lg);
            // wave32 reduction over 32 of the 64 d-lanes, then 2 LDS atomics/j
            for (int off = 16; off > 0; off >>= 1)
                mah += __shfl_down(mah, off, 32);
            if ((t & 31) == 0) atomicAdd(&dist[j], mah);
        } else {
            // write-once zeros, never re-read: keep them out of L2's way
            __builtin_nontemporal_store(0.0f, &out[zbase_r + (size_t)j * DV_ + d]);
            __builtin_nontemporal_store(0.0f, &out[zbase_i + (size_t)j * DV_ + d]);
            __builtin_nontemporal_store(0.0f, &out[qbase   + (size_t)j * DV_ + d]);
        }
    }
    __syncthreads();

    // Stage B: Vij -> scores, in place
    for (int j = jg; j <= i; j += 4) {
        float Vij  = sc[j * DV_ + d];
        float base = Vij * (nf + nu_abs * dist[j] + EPS);
        sc[j * DV_ + d] = -tau_abs * __logf(base);
    }
    __syncthreads();

    // Stage C: per-d softmax stats over j <= i
    if (t < DV_) {
        float m = -3.402823466e38f;
        for (int j = 0; j <= i; ++j) m = fmaxf(m, sc[j * DV_ + t]);
        float s = 0.0f;
        for (int j = 0; j <= i; ++j) s += __expf(sc[j * DV_ + t] - m);
        dmax[t] = m; dsum[t] = s;
    }
    __syncthreads();

    // Stage D: Qij writes + est_v accumulation. Zhat is re-read from L2
    // (same-wave store->load to identical addresses is kept in order).
    float md = dmax[d];
    float rs = 1.0f / dsum[d];
    float er = 0.0f, ei = 0.0f;
    for (int j = jg; j <= i; j += 4) {
        float q = __expf(sc[j * DV_ + d] - md) * rs;
        __builtin_nontemporal_store(q, &out[qbase + (size_t)j * DV_ + d]);  // never re-read
        float zr = out[zbase_r + (size_t)j * DV_ + d];
        float zi = out[zbase_i + (size_t)j * DV_ + d];
        er += q * zr;
        ei += q * zi;
    }
    atomicAdd(&estv[d], er);
    atomicAdd(&estv[DV_ + d], ei);
    __syncthreads();

    // eta-gate, write est_latent
    if (t < DV_) {
        float estr = estv[t], esti = estv[DV_ + t];
        float4 kvI = kv4[(b * 256 + i) * DV_ + t];
        float vr = kvI.z, vi = kvI.w;
        float eta = 1.0f / (1.0f + __expf(-eta_p[t]));
        float g   = 1.0f / (1.0f + __expf(-eta));
        out[OFF_EST + ((b * 2 + 0) * N_ + i) * DV_ + t] = (1.0f - eta) * vr + g * estr;
        out[OFF_EST + ((b * 2 + 1) * N_ + i) * DV_ + t] = (1.0f - eta) * vi + g * esti;
    }
}

// ---------------------------------------------------------------------------
// Kernel 3: one-step propagation + complex output projection via WMMA.
// (1024 x 64) @ (64 x 128); A fetched on the fly from est_latent with the
// per-channel propagator (mr, mi). 64 blocks * 8 waves = 512 tiles.
// ---------------------------------------------------------------------------
__global__ __launch_bounds__(256) void out_proj_kernel(
    const float* __restrict__ T, const float* __restrict__ lam1,
    const float* __restrict__ Wpr, const float* __restrict__ Wpi,
    const float* __restrict__ bpr, const float* __restrict__ bpi,
    float* __restrict__ out)
{
    const float* est = out + OFF_EST;
    float h = T[1] - T[0];
    int wave = threadIdx.x >> 5, lane = threadIdx.x & 31;
    int gid = blockIdx.x * 8 + wave;
    int mt = gid >> 3, nt = gid & 7;
    int mloc = lane & 15, khalf = lane >> 4;
    int row_m = mt * 16 + mloc;
    int rb = row_m >> 8, rn = row_m & 255;
    int elr_base = (rb * 512 + rn) * DV_;
    int eli_base = elr_base + 256 * DV_;
    int col = nt * 16 + mloc;

    v8f arr = {}; v8f aii = {}; v8f ari = {}; v8f air = {};
    for (int k = 0; k < DV_; k += 4) {
        int k0 = k + 2 * khalf;
        v2f ppr, ppi, bwr, bwi;
#pragma unroll
        for (int u = 0; u < 2; ++u) {
            int kk = k0 + u;
            float l0 = lam1[kk & 31];
            float re = -l0 * l0;
            float l1 = lam1[32 + (kk & 31)];
            float im = (kk < 32) ? l1 : -l1;
            float ma = __expf(re * h);
            float s, c; __sincosf(im * h, &s, &c);
            float mr = ma * c, mi = ma * s;
            float er = est[elr_base + kk], ei = est[eli_base + kk];
            float pr = mr * er - mi * ei;
            float pi = mr * ei + mi * er;
            float wr = Wpr[kk * DE_ + col], wi = Wpi[kk * DE_ + col];
            if (u == 0) { ppr.x = pr; ppi.x = pi; bwr.x = wr; bwi.x = wi; }
            else        { ppr.y = pr; ppi.y = pi; bwr.y = wr; bwi.y = wi; }
        }
        arr = __builtin_amdgcn_wmma_f32_16x16x4_f32(false, ppr, false, bwr, (short)0, arr, false, false);
        aii = __builtin_amdgcn_wmma_f32_16x16x4_f32(false, ppi, false, bwi, (short)0, aii, false, false);
        ari = __builtin_amdgcn_wmma_f32_16x16x4_f32(false, ppr, false, bwi, (short)0, ari, false, false);
        air = __builtin_amdgcn_wmma_f32_16x16x4_f32(false, ppi, false, bwr, (short)0, air, false, false);
    }
    float brv = bpr[col], biv = bpi[col];
#pragma unroll
    for (int v = 0; v < 8; ++v) {
        int row = mt * 16 + v + 8 * khalf;
        int b = row >> 8, n = row & 255;
        size_t obase = (size_t)OFF_OUT + (size_t)(b * 512 + n) * DE_ + col;
        out[obase]             = arr[v] - aii[v] + brv;
        out[obase + 256 * DE_] = ari[v] + air[v] + biv;
    }
}

// ---------------------------------------------------------------------------
extern "C" void kernel_launch(void* const* d_in, const int* in_sizes, int n_in,
                              void* d_out, int out_size, void* d_ws, size_t ws_size,
                              hipStream_t stream) {
    const float* Zq  = (const float*)d_in[0];
    const float* Zk  = (const float*)d_in[1];
    const float* Zv  = (const float*)d_in[2];
    const float* T   = (const float*)d_in[3];
    const float* Wqr = (const float*)d_in[4];
    const float* Wqi = (const float*)d_in[5];
    const float* bqr = (const float*)d_in[6];
    const float* bqi = (const float*)d_in[7];
    const float* Wkr = (const float*)d_in[8];
    const float* Wki = (const float*)d_in[9];
    const float* bkr = (const float*)d_in[10];
    const float* bki = (const float*)d_in[11];
    const float* Wvr = (const float*)d_in[12];
    const float* Wvi = (const float*)d_in[13];
    const float* bvr = (const float*)d_in[14];
    const float* bvi = (const float*)d_in[15];
    const float* Wpr = (const float*)d_in[16];
    const float* Wpi = (const float*)d_in[17];
    const float* bpr = (const float*)d_in[18];
    const float* bpi = (const float*)d_in[19];
    const float* lam1 = (const float*)d_in[20];
    const float* lOm  = (const float*)d_in[21];
    const float* lGa  = (const float*)d_in[22];
    const float* nf   = (const float*)d_in[23];
    const float* tau  = (const float*)d_in[24];
    const float* nu   = (const float*)d_in[25];
    const float* etap = (const float*)d_in[26];

    float* ws  = (float*)d_ws;   // Qr, Qi, packed KV : 393216 floats = 1.5 MB
    float* out = (float*)d_out;

    proj_qkv_kernel<<<96, 256, 0, stream>>>(Zq, Zk, Zv,
        Wqr, Wqi, bqr, bqi, Wkr, Wki, bkr, bki, Wvr, Wvi, bvr, bvi,
        lam1, ws, out);

    size_t smem = (size_t)(N_ * DV_ + N_ + DV_ + DV_ + 2 * DV_) * sizeof(float); // 67,584 B
    attn_kernel<<<B_ * N_, 256, smem, stream>>>(T, lam1, lOm, lGa, nf, tau, nu, etap, ws, out);

    out_proj_kernel<<<64, 256, 0, stream>>>(T, lam1, Wpr, Wpi, bpr, bpi, out);
}